// S4DKernel_64175401337549
// MI455X (gfx1250) — compile-verified
//
#include <hip/hip_runtime.h>
#include <hip/hip_bf16.h>

typedef float v2f __attribute__((ext_vector_type(2)));
typedef float v8f __attribute__((ext_vector_type(8)));

#define D_MODEL 1024
#define N_STATE 64
#define SEQ_LEN 1024
#define QPAD 72   // 2*72 = 144 ≡ 16 (mod 64): hi/lo half-wave A-operand reads hit disjoint banks

// One workgroup (4 wave32s) per channel d.
// K[d, 16q+r] = Re( sum_n u[n,q] * v[n,r] ),  u = dt*c*E^(16q), v = E^r, E = exp(dtA)
//             = Ur^T Vr + (-Ui)^T Vi   ->  two M64xN16xK64 f32 matmuls via v_wmma_f32_16x16x4_f32
// Powers of E built from only 3 precise exp/sincos seeds per n (E^1, E^16, E^128)
// via short complex-multiply recurrences (max 15 steps -> ~2e-6 rel. error).
__global__ __launch_bounds__(128) void s4d_vandermonde_wmma(
    const float* __restrict__ A_real, const float* __restrict__ A_imag,
    const float* __restrict__ Cm, const float* __restrict__ log_dt,
    const float* __restrict__ Bm, float* __restrict__ out)
{
    __shared__ float s_ur  [N_STATE][QPAD];  // Re(u)[n][q]
    __shared__ float s_uin [N_STATE][QPAD];  // -Im(u)[n][q]
    __shared__ float s_vr  [N_STATE][16];    // Re(E^r)
    __shared__ float s_vi  [N_STATE][16];    // Im(E^r)
    __shared__ float s_p16r[N_STATE][8];     // Re(E^(16*ql)), ql=0..7
    __shared__ float s_p16i[N_STATE][8];
    __shared__ float s_p128r[N_STATE][8];    // Re(E^(128*qh)), qh=0..7
    __shared__ float s_p128i[N_STATE][8];
    __shared__ float s_cr[N_STATE], s_ci[N_STATE];  // dt * (C0,C1) * B

    const int d   = blockIdx.x;
    const int tid = threadIdx.x;
    const float dt = expf(log_dt[d]);

    // ---- Phase 1a: seed powers of E = exp(dtA) with 3 transcendental pairs per n ----
    {
        const int n = tid & (N_STATE - 1);
        const float dar = dt * A_real[d * N_STATE + n];
        const float dai = dt * A_imag[d * N_STATE + n];
        if (tid < N_STATE) {
            const float b = Bm[d * N_STATE + n];
            s_cr[n] = dt * Cm[(d * N_STATE + n) * 2 + 0] * b;
            s_ci[n] = dt * Cm[(d * N_STATE + n) * 2 + 1] * b;
            // v chain: E^r, r = 0..15, seeded by E^1
            float e = expf(dar);
            float s, c;
            sincosf(dai, &s, &c);
            const float er = e * c, ei = e * s;
            float wr = 1.0f, wi = 0.0f;
            s_vr[n][0] = 1.0f; s_vi[n][0] = 0.0f;
            #pragma unroll
            for (int r = 1; r < 16; ++r) {
                const float nr = wr * er - wi * ei;
                const float ni = wr * ei + wi * er;
                wr = nr; wi = ni;
                s_vr[n][r] = wr; s_vi[n][r] = wi;
            }
        } else {
            // p16 chain: E^(16*ql), seeded by E^16;  p128 chain: E^(128*qh), seeded by E^128
            float e16 = expf(16.0f * dar);
            float s16, c16;
            sincosf(16.0f * dai, &s16, &c16);
            const float er16 = e16 * c16, ei16 = e16 * s16;
            float e128 = expf(128.0f * dar);
            float s128, c128;
            sincosf(128.0f * dai, &s128, &c128);
            const float er128 = e128 * c128, ei128 = e128 * s128;

            float wr = 1.0f, wi = 0.0f;
            s_p16r[n][0] = 1.0f; s_p16i[n][0] = 0.0f;
            #pragma unroll
            for (int j = 1; j < 8; ++j) {
                const float nr = wr * er16 - wi * ei16;
                const float ni = wr * ei16 + wi * er16;
                wr = nr; wi = ni;
                s_p16r[n][j] = wr; s_p16i[n][j] = wi;
            }
            wr = 1.0f; wi = 0.0f;
            s_p128r[n][0] = 1.0f; s_p128i[n][0] = 0.0f;
            #pragma unroll
            for (int j = 1; j < 8; ++j) {
                const float nr = wr * er128 - wi * ei128;
                const float ni = wr * ei128 + wi * er128;
                wr = nr; wi = ni;
                s_p128r[n][j] = wr; s_p128i[n][j] = wi;
            }
        }
    }
    __syncthreads();

    // ---- Phase 1b: u[n,q] = (dt*c_n) * E^(16*ql) * E^(128*qh),  q = 8*qh + ql ----
    for (int idx = tid; idx < N_STATE * 64; idx += 128) {
        const int n = idx >> 6, q = idx & 63;
        const int ql = q & 7, qh = q >> 3;
        const float ar = s_p16r[n][ql],  ai = s_p16i[n][ql];
        const float br = s_p128r[n][qh], bi = s_p128i[n][qh];
        const float wr = ar * br - ai * bi;
        const float wi = ar * bi + ai * br;
        const float cr = s_cr[n], ci = s_ci[n];
        s_ur [n][q] =  cr * wr - ci * wi;
        s_uin[n][q] = -(cr * wi + ci * wr);
    }
    __syncthreads();

    // ---- Phase 2: wave w computes 16x16 output tile (q in [16w,16w+16), r in [0,16)) ----
    const int wave = tid >> 5;
    const int lane = tid & 31;
    const int m    = lane & 15;
    const int hi   = lane >> 4;        // 0: K=0,1 halves; 1: K=2,3 halves
    const int q0   = wave * 16;

    v8f acc_r = {};
    v8f acc_i = {};
    #pragma unroll
    for (int k = 0; k < N_STATE; k += 4) {
        const int ka = k + (hi ? 2 : 0);
        v2f a_r, a_i, b_r, b_i;
        // A (16x4, M=q-local, K=n): lane holds [M=m][K=ka], [M=m][K=ka+1]
        a_r[0] = s_ur [ka][q0 + m];  a_r[1] = s_ur [ka + 1][q0 + m];
        a_i[0] = s_uin[ka][q0 + m];  a_i[1] = s_uin[ka + 1][q0 + m];
        // B (4x16, K=n, N=r): lane holds [K=ka][N=m], [K=ka+1][N=m]
        b_r[0] = s_vr[ka][m];        b_r[1] = s_vr[ka + 1][m];
        b_i[0] = s_vi[ka][m];        b_i[1] = s_vi[ka + 1][m];

        acc_r = __builtin_amdgcn_wmma_f32_16x16x4_f32(
            false, a_r, false, b_r, (short)0, acc_r, false, false);
        acc_i = __builtin_amdgcn_wmma_f32_16x16x4_f32(
            false, a_i, false, b_i, (short)0, acc_i, false, false);
    }
    const v8f acc = acc_r + acc_i;

    // D layout: VGPR j -> M = j (lanes 0-15) / j+8 (lanes 16-31), N = m
    float* outd = out + (size_t)d * SEQ_LEN;
    #pragma unroll
    for (int j = 0; j < 8; ++j) {
        const int ql2 = j + (hi ? 8 : 0);        // q within tile
        outd[(q0 + ql2) * 16 + m] = acc[j];      // l = 16*(q0+ql2) + r
    }
}

extern "C" void kernel_launch(void* const* d_in, const int* in_sizes, int n_in,
                              void* d_out, int out_size, void* d_ws, size_t ws_size,
                              hipStream_t stream) {
    const float* A_real = (const float*)d_in[0];
    const float* A_imag = (const float*)d_in[1];
    const float* C      = (const float*)d_in[2];
    const float* log_dt = (const float*)d_in[3];
    const float* B      = (const float*)d_in[4];
    // d_in[5] is L (==1024), baked into the tiling (SEQ_LEN = 64 q-tiles * 16 r)
    float* out = (float*)d_out;

    dim3 grid(D_MODEL), block(128);
    hipLaunchKernelGGL(s4d_vandermonde_wmma, grid, block, 0, stream,
                       A_real, A_imag, C, log_dt, B, out);
}